// DCT8x8_14439680049297
// MI455X (gfx1250) — compile-verified
//
#include <hip/hip_runtime.h>
#include <math.h>
#include <stdint.h>

// ---- vector types ----
typedef float        v2f  __attribute__((ext_vector_type(2)));
typedef float        v4f  __attribute__((ext_vector_type(4)));
typedef float        v8f  __attribute__((ext_vector_type(8)));
typedef unsigned int v4u  __attribute__((ext_vector_type(4)));
typedef int          v4i  __attribute__((ext_vector_type(4)));
typedef int          v8i  __attribute__((ext_vector_type(8)));

#define WAVES_PER_BLOCK 2
// Per-wave LDS dword budgets
#define IN_DW   2688   // 16 rows * 8 segments * (16 data + 5 pad) dwords  (TDM-padded input strip)
#define TB_DW   320    // 16 x 20 transpose buffer (stride 20 -> conflict-free)
#define STG_DW  2048   // 128 rows x 16 dwords output staging (XOR swizzled)
#define WAVE_DW (IN_DW + TB_DW + STG_DW)   // 5056 dwords = 20224 B / wave

__global__ __launch_bounds__(WAVES_PER_BLOCK * 32)
void DCT8x8_cdna5_kernel(const float* __restrict__ x, const float* __restrict__ dct,
                         float* __restrict__ out) {
  __shared__ float smem[WAVE_DW * WAVES_PER_BLOCK];

  const int lane = threadIdx.x & 31;
  const int wave = threadIdx.x >> 5;
  const int gid  = blockIdx.x * WAVES_PER_BLOCK + wave;   // 0 .. 32767, exact fit

  const int b  = gid >> 9;       // batch 0..63
  const int r9 = gid & 511;
  const int sr = r9 >> 3;        // strip row 0..63  -> image rows sr*16 .. sr*16+15
  const int sc = r9 & 7;         // column group 0..7 -> image cols sc*128 .. sc*128+127

  float* inb = smem + wave * WAVE_DW;   // TDM-staged input strip
  float* tb  = inb + IN_DW;             // per-tile transpose buffer
  float* stg = tb + TB_DW;              // output staging

  const int lo = lane & 15;             // 0..15
  const int hi = lane >> 4;             // 0 / 1 (half-wave)

  // ------------------------------------------------------------------
  // Constant A operand: 16x4 column slices of Md = diag(M, M), M = the 8x8
  // DCT matrix supplied by the harness (bit-identical to the reference).
  // A layout (f32 16x16x4): lanes 0-15 hold row=lane, K = kb,kb+1 (v0,v1);
  // lanes 16-31 hold row=lane-16, K = kb+2,kb+3.  Same A used by BOTH stages.
  // Chunk kb4 uses Md columns c0 = 4*kb4 + 2*hi; (c0 & 7) only takes the two
  // values {2*hi, 4+2*hi}, so two b64 loads of M cover all four chunks; the
  // block-diagonal structure is applied with selects.
  // ------------------------------------------------------------------
  const int mu   = lo & 7;              // DCT index u of this lane's Md row
  const int mblk = lo >> 3;             // which diagonal 8x8 block (row side)
  const v2f pairLo = *(const v2f*)(dct + mu * 8 + 2 * hi);       // M[mu][2hi..2hi+1]
  const v2f pairHi = *(const v2f*)(dct + mu * 8 + 4 + 2 * hi);   // M[mu][4+2hi..5+2hi]
  const v2f zero2  = (v2f){0.f, 0.f};
  v2f amat[4];
  amat[0] = (mblk == 0) ? pairLo : zero2;   // c0 = 0..7   -> left diagonal block
  amat[1] = (mblk == 0) ? pairHi : zero2;
  amat[2] = (mblk == 1) ? pairLo : zero2;   // c0 = 8..15  -> right diagonal block
  amat[3] = (mblk == 1) ? pairHi : zero2;

  // ------------------------------------------------------------------
  // TDM: async-load the 16x128 fp32 strip global -> LDS with padding:
  // 5 pad dwords after every 16 data dwords, so element (r, 16t+n) lands at
  // dword offset 168*r + 21*t + n   (conflict-free B-operand reads).
  // ------------------------------------------------------------------
  const unsigned long long gaddr = (unsigned long long)(uintptr_t)(
      x + ((size_t)b << 20) + ((size_t)(sr * 16) << 10) + (size_t)sc * 128);
  const unsigned int ldsoff = (unsigned int)(uintptr_t)inb;  // low 32 bits = LDS byte offset

  v4u g0;
  g0.x = 1u;                                                       // count=1, user mode
  g0.y = (unsigned int)__builtin_amdgcn_readfirstlane((int)ldsoff);                 // lds_addr
  g0.z = (unsigned int)__builtin_amdgcn_readfirstlane((int)(gaddr & 0xffffffffu));  // global lo
  g0.w = ((unsigned int)__builtin_amdgcn_readfirstlane((int)((gaddr >> 32) & 0x01ffffffu)))
         | 0x80000000u;                                            // global hi | type=2
  v8i g1;
  g1[0] = (int)((2u << 16)      // data_size = 4 bytes
              | (1u << 20)      // pad_enable
              | (3u << 22)      // pad_interval: 16 dwords
              | (4u << 25));    // pad_amount: 5 dwords
  g1[1] = (int)((1024u & 0xffffu) << 16);                       // tensor_dim0 lo16 (=1024)
  g1[2] = (int)(((1024u >> 16) & 0xffffu) | ((1024u & 0xffffu) << 16)); // dim0 hi | dim1 lo (=1024)
  g1[3] = (int)(((1024u >> 16) & 0xffffu) | (128u << 16));      // dim1 hi | tile_dim0 = 128
  g1[4] = 16;                                                   // tile_dim1 = 16, tile_dim2 = 0
  g1[5] = 1024;                                                 // tensor_dim0_stride lo32
  g1[6] = 0;
  g1[7] = 0;
  v4i g2 = (v4i){0, 0, 0, 0};
  v4i g3 = (v4i){0, 0, 0, 0};
  v8i g4 = (v8i){0, 0, 0, 0, 0, 0, 0, 0};   // extra group on 6-arg toolchain: zero/no-op
  __builtin_amdgcn_tensor_load_to_lds(g0, g1, g2, g3, g4, 0);
  __asm__ volatile("s_wait_tensorcnt 0" ::: "memory");

  // ------------------------------------------------------------------
  // Process the 8 16x16 tiles of the strip.
  // ------------------------------------------------------------------
#pragma unroll
  for (int t = 0; t < 8; ++t) {
    const float* tbase = inb + t * 21;

    // Stage 1: T = Md * X   (B operand: X rows kb..kb+3, one column per lane)
    v8f acc = (v8f){0.f, 0.f, 0.f, 0.f, 0.f, 0.f, 0.f, 0.f};
#pragma unroll
    for (int kb4 = 0; kb4 < 4; ++kb4) {
      const int r0 = kb4 * 4 + 2 * hi;
      v2f bm;
      bm.x = tbase[(r0)*168 + lo];
      bm.y = tbase[(r0 + 1) * 168 + lo];
      acc = __builtin_amdgcn_wmma_f32_16x16x4_f32(false, amat[kb4], false, bm,
                                                  (short)0, acc, false, false);
    }

    // Transpose T through LDS (row stride 20 dwords -> conflict-free both ways)
#pragma unroll
    for (int v = 0; v < 8; ++v) {
      tb[(v + 8 * hi) * 20 + lo] = acc[v];
    }
    __asm__ volatile("s_wait_dscnt 0" ::: "memory");

    // Stage 2: Z = Md * T^T  (= Y^T).  B operand rows kb..kb+3 of T^T are
    // column pairs of T: one ds_load_b64 per chunk per lane.
    v8f acc2 = (v8f){0.f, 0.f, 0.f, 0.f, 0.f, 0.f, 0.f, 0.f};
#pragma unroll
    for (int kb4 = 0; kb4 < 4; ++kb4) {
      const int c0 = kb4 * 4 + 2 * hi;
      const v2f bm = *(const v2f*)(tb + lo * 20 + c0);
      acc2 = __builtin_amdgcn_wmma_f32_16x16x4_f32(false, amat[kb4], false, bm,
                                                   (short)0, acc2, false, false);
    }
    __asm__ volatile("s_wait_dscnt 0" ::: "memory");

    // Stage Z into the output-staging buffer.
    // acc2[v] = Y[lo, v + 8*hi];  Y row = 8*di + u, Y col = 8*dj + v.
    // Staging layout: logical [cc][i2][j16] row R = 2*cc + i2, 16 dwords/row,
    // XOR-swizzled in 4-dword groups (keeps b128 alignment, 2-way conflicts max).
#pragma unroll
    for (int v = 0; v < 8; ++v) {
      const int u  = lo & 7;
      const int di = lo >> 3;
      const int cc = u * 8 + v;            // output channel = u*8 + v
      const int R  = cc * 2 + di;
      const int c  = t * 2 + hi;           // j within strip: 0..15
      const int c4 = c >> 2, c2 = c & 3;
      const int sw = (c4 ^ (R & 3) ^ ((R >> 4) & 3)) & 3;
      stg[R * 16 + sw * 4 + c2] = acc2[v];
    }
  }

  // ------------------------------------------------------------------
  // Coalesced writeback: 4 lanes cover one 64-byte channel-row (16 floats),
  // stored as non-temporal global_store_b128 (output is stream-once).
  // out[b, cc, i, j], strides (64*128*128, 128*128, 128, 1).
  // ------------------------------------------------------------------
  __asm__ volatile("s_wait_dscnt 0" ::: "memory");
  float* outb = out + ((size_t)b << 20);
  const int q  = lane >> 2;   // 0..7: row within pass
  const int c4 = lane & 3;    // 4-dword group within row
#pragma unroll
  for (int p = 0; p < 16; ++p) {
    const int R  = p * 8 + q;            // 0..127
    const int cc = R >> 1;
    const int i2 = R & 1;
    const int sw = (c4 ^ (R & 3) ^ ((R >> 4) & 3)) & 3;
    const v4f val = *(const v4f*)(stg + R * 16 + sw * 4);
    const int i = sr * 2 + i2;           // block row 0..127
    const int j = sc * 16 + c4 * 4;      // block col, 16B aligned
    __builtin_nontemporal_store(val, (v4f*)(outb + ((size_t)cc << 14) + (size_t)i * 128 + j));
  }
}

extern "C" void kernel_launch(void* const* d_in, const int* in_sizes, int n_in,
                              void* d_out, int out_size, void* d_ws, size_t ws_size,
                              hipStream_t stream) {
  (void)in_sizes; (void)n_in; (void)d_ws; (void)ws_size; (void)out_size;
  const float* x   = (const float*)d_in[0];   // [64,1,1024,1024] fp32
  const float* dct = (const float*)d_in[1];   // [8,8] fp32 DCT-II matrix
  float* out = (float*)d_out;                 // [64,64,128,128] fp32
  // 32768 strips (16 rows x 128 cols each), 2 waves per 64-thread block.
  DCT8x8_cdna5_kernel<<<dim3(32768 / WAVES_PER_BLOCK), dim3(WAVES_PER_BLOCK * 32), 0, stream>>>(x, dct, out);
}